// CachedConv2d_63153199120698
// MI455X (gfx1250) — compile-verified
//
#include <hip/hip_runtime.h>

typedef __attribute__((ext_vector_type(2))) float v2f;
typedef __attribute__((ext_vector_type(8))) float v8f;

#define CIN    32
#define COUT   32
#define HH     256
#define WW     256
#define TILE_W 128
#define NSTEP  72              // K=288 in steps of 4

// LDS layout (in floats)
#define XPADW   34             // 32 ci + 2 pad (even -> 8B-aligned pairs, conflict-free stride)
#define XROWW   130            // 128 w + 2 halo
#define X_ELEMS (3 * XROWW * XPADW)        // 13260 (even)
#define W_OFF   X_ELEMS
#define W_FRAGS (NSTEP * 2 * 32)           // 4608 float2 fragments
#define B_OFF   (W_OFF + W_FRAGS * 2)      // 22476
#define SMEM_FLOATS (B_OFF + 32)           // 22508 floats = 90032 bytes

extern "C" __global__ __launch_bounds__(256, 2)
void conv3x3_wmma_f32(const float* __restrict__ x,
                      const float* __restrict__ wgt,
                      const float* __restrict__ bias,
                      float* __restrict__ out)
{
    extern __shared__ float smem[];
    float* xlds = smem;
    v2f*   wlds = (v2f*)(smem + W_OFF);
    float* blds = smem + B_OFF;

    const int tid   = threadIdx.x;
    const int blk   = blockIdx.x;
    const int b     = blk >> 9;          // 512 blocks per batch image
    const int h     = (blk >> 1) & 255;
    const int wtile = blk & 1;
    const int wbase_g = wtile * TILE_W;

    // ---- stage weights as exact WMMA A-fragments (cout x k), k = (kh*3+kw)*32 + ci ----
    for (int idx = tid; idx < W_FRAGS; idx += 256) {
        int s    = idx >> 6;             // k-step
        int rem  = idx & 63;
        int t    = rem >> 5;             // cout tile (0: 0-15, 1: 16-31)
        int l    = rem & 31;             // lane
        int k0   = 4 * s + ((l >> 4) << 1);
        int cout = (t << 4) + (l & 15);
        v2f wv;
        #pragma unroll
        for (int v = 0; v < 2; ++v) {
            int k    = k0 + v;
            int ci   = k & 31;
            int khkw = k >> 5;           // == kh*3+kw
            wv[v] = wgt[(cout * CIN + ci) * 9 + khkw];
        }
        wlds[idx] = wv;
    }

    // ---- stage bias ----
    if (tid < 32) blds[tid] = bias[tid];

    // ---- stage 3 input rows (h-1,h,h+1), channel-interleaved with zero halo ----
    for (int idx = tid; idx < 3 * CIN * XROWW; idx += 256) {
        int kh = idx / (CIN * XROWW);
        int r  = idx - kh * (CIN * XROWW);
        int ci = r / XROWW;
        int wp = r - ci * XROWW;
        int hin = h + kh - 1;
        int win = wbase_g + wp - 1;
        float v = 0.0f;
        if ((unsigned)hin < HH && (unsigned)win < WW)
            v = x[((b * CIN + ci) * HH + hin) * WW + win];
        xlds[(kh * XROWW + wp) * XPADW + ci] = v;
    }
    __syncthreads();

    // ---- implicit-GEMM compute: each wave -> 32(cout) x 16(w) output tile ----
    const int lane = tid & 31;
    const int wl   = (tid >> 5) << 4;    // local w base: 0..112
    const int nco  = lane & 15;          // N column within tile (w offset)
    const int hi   = lane >> 4;          // lane half select

    v8f acc0 = {};                       // cout 0..15
    v8f acc1 = {};                       // cout 16..31

    #pragma unroll
    for (int s = 0; s < NSTEP; ++s) {
        const int khkw = s >> 3;         // constant after unroll
        const int kh   = khkw / 3;
        const int kw   = khkw - 3 * kh;
        const int cig  = (s & 7) << 2;   // ci group base

        v2f a0 = wlds[(s * 2 + 0) * 32 + lane];
        v2f a1 = wlds[(s * 2 + 1) * 32 + lane];

        const int wp = wl + nco + kw;            // includes +1 halo shift (kw-1 + 1)
        const int ci = cig + (hi << 1);          // even -> 8B aligned pair
        v2f bv = *(const v2f*)&xlds[(kh * XROWW + wp) * XPADW + ci];

        acc0 = __builtin_amdgcn_wmma_f32_16x16x4_f32(
                   false, a0, false, bv, (short)0, acc0, false, false);
        acc1 = __builtin_amdgcn_wmma_f32_16x16x4_f32(
                   false, a1, false, bv, (short)0, acc1, false, false);
    }

    // ---- epilogue: bias add + coalesced stores (16 contiguous w per half-wave) ----
    const int wcol = wbase_g + wl + nco;
    #pragma unroll
    for (int r = 0; r < 8; ++r) {
        int co = r + (hi << 3);
        float v0 = acc0[r] + blds[co];
        float v1 = acc1[r] + blds[co + 16];
        out[((b * COUT + co)      * HH + h) * WW + wcol] = v0;
        out[((b * COUT + co + 16) * HH + h) * WW + wcol] = v1;
    }
}

extern "C" void kernel_launch(void* const* d_in, const int* in_sizes, int n_in,
                              void* d_out, int out_size, void* d_ws, size_t ws_size,
                              hipStream_t stream) {
    const float* x    = (const float*)d_in[0];
    const float* wgt  = (const float*)d_in[1];
    const float* bias = (const float*)d_in[2];
    float* out = (float*)d_out;

    dim3 grid(16 * 256 * 2);   // B * H * (W / TILE_W)
    dim3 block(256);           // 8 wave32
    size_t smem = SMEM_FLOATS * sizeof(float);   // ~90 KB of the 320 KB WGP LDS
    hipLaunchKernelGGL(conv3x3_wmma_f32, grid, block, smem, stream, x, wgt, bias, out);
}